// HashGrid_79542794322241
// MI455X (gfx1250) — compile-verified
//
#include <hip/hip_runtime.h>
#include <stdint.h>

// HashGrid (Instant-NGP style) forward encode for MI455X / gfx1250.
//
// Thread mapping: tid -> (point = tid>>4, level = tid&15). One wave32 covers
// two points (16 levels each). Per thread: 16 independent 4B gathers from the
// L2-resident 64MB table, trilinear reduce, one coalesced nontemporal float2
// store. Memory-bound on L2 gather bandwidth; WMMA intentionally unused (no
// shared-operand matrix product exists in this workload).

typedef __attribute__((ext_vector_type(2))) float v2f;

namespace {
constexpr int      kL     = 16;
constexpr int      kF     = 2;
constexpr uint32_t kT     = 1u << 19;
constexpr uint32_t kTMask = kT - 1u;
constexpr uint32_t kP1    = 2654435761u;   // PRIMES[1]
constexpr uint32_t kP2    = 805459861u;    // PRIMES[2]  (PRIMES[0] == 1)
}

__global__ __launch_bounds__(256)
void hashgrid_fwd_kernel(const float* __restrict__ x,
                         const float* __restrict__ hashmap,
                         const float* __restrict__ resolution,
                         float* __restrict__ out,
                         int n)
{
    const uint32_t tid   = blockIdx.x * blockDim.x + threadIdx.x;
    const uint32_t level = tid & (kL - 1);      // lane % 16
    const uint32_t point = tid >> 4;
    if (point >= (uint32_t)n) return;

    // Per-lane scale for this level (16-entry table, L0-cached broadcast).
    const float res = resolution[level];

    // All 16 lanes of a half-wave read the same 12 bytes; VMEM coalesces
    // same-address requests, so this is one L0 transaction per point.
    const float px = x[point * 3 + 0];
    const float py = x[point * 3 + 1];
    const float pz = x[point * 3 + 2];

    // Match reference math exactly: f32 scale (rounded once), floor, frac.
    float sx = px * res, sy = py * res, sz = pz * res;
    // Materialize the rounded products so the compiler cannot re-fuse the
    // frac subtraction into v_fma_f32 (frac must be round(x*res) - floor,
    // bit-identical to the reference, not fma(x, res, -floor)).
    asm volatile("" : "+v"(sx), "+v"(sy), "+v"(sz));
    const float fx0 = floorf(sx), fy0 = floorf(sy), fz0 = floorf(sz);
    const float fx = sx - fx0, fy = sy - fy0, fz = sz - fz0;
    const uint32_t ix = (uint32_t)fx0, iy = (uint32_t)fy0, iz = (uint32_t)fz0;

    // Per-dimension hashes for corner offsets {0,1} (uint32 wraparound mul).
    const uint32_t hx[2] = { ix,         ix + 1u           };
    const uint32_t hy[2] = { iy * kP1,   (iy + 1u) * kP1   };
    const uint32_t hz[2] = { iz * kP2,   (iz + 1u) * kP2   };
    const float    wx[2] = { 1.0f - fx,  fx };
    const float    wy[2] = { 1.0f - fy,  fy };
    const float    wz[2] = { 1.0f - fz,  fz };

    // hashmap layout [L, F, T]: feature slabs for this level are T apart.
    const float* __restrict__ b0 = hashmap + (size_t)level * (size_t)(kF) * (size_t)kT;
    const float* __restrict__ b1 = b0 + kT;

    // Phase 1: issue all 16 gathers back-to-back (fully unrolled) so the
    // wave has maximum LOADcnt in flight before the first dependent use.
    float f0[8], f1[8], w[8];
    #pragma unroll
    for (int c = 0; c < 8; ++c) {
        const int dx = (c >> 2) & 1;
        const int dy = (c >> 1) & 1;
        const int dz =  c       & 1;
        const uint32_t id = (hx[dx] ^ hy[dy] ^ hz[dz]) & kTMask;
        w[c]  = wx[dx] * wy[dy] * wz[dz];
        f0[c] = b0[id];
        f1[c] = b1[id];
    }

    // Phase 2: trilinear weighted reduce (sequential FMA).
    float acc0 = 0.0f, acc1 = 0.0f;
    #pragma unroll
    for (int c = 0; c < 8; ++c) {
        acc0 = fmaf(w[c], f0[c], acc0);
        acc1 = fmaf(w[c], f1[c], acc1);
    }

    // Output (n, L*F): this thread owns out[point, level*2 .. level*2+1].
    // Consecutive lanes -> consecutive float2 -> 256B coalesced wave store.
    // Non-temporal: the 64MB output is write-once; keep it from evicting the
    // L2-resident hash table (the thing this kernel's perf depends on).
    v2f o;
    o.x = acc0;
    o.y = acc1;
    v2f* dst = (v2f*)(out + (size_t)point * (size_t)(kL * kF) + (size_t)level * 2u);
    __builtin_nontemporal_store(o, dst);
}

extern "C" void kernel_launch(void* const* d_in, const int* in_sizes, int n_in,
                              void* d_out, int out_size, void* d_ws, size_t ws_size,
                              hipStream_t stream) {
    (void)n_in; (void)out_size; (void)d_ws; (void)ws_size;
    const float* x          = (const float*)d_in[0];   // [n, 3]
    const float* hashmap    = (const float*)d_in[1];   // [16, 2, 2^19]
    const float* resolution = (const float*)d_in[2];   // [16]
    float*       out        = (float*)d_out;           // [n, 32]

    const int n       = in_sizes[0] / 3;
    const int threads = 256;                 // 8 wave32s per block
    const long total  = (long)n * kL;        // one thread per (point, level)
    const int  blocks = (int)((total + threads - 1) / threads);

    hipLaunchKernelGGL(hashgrid_fwd_kernel, dim3(blocks), dim3(threads), 0, stream,
                       x, hashmap, resolution, out, n);
}